// base_GC_net_49151605735515
// MI455X (gfx1250) — compile-verified
//
#include <hip/hip_runtime.h>
#include <hip/hip_bf16.h>

typedef __attribute__((ext_vector_type(2))) float v2f;
typedef __attribute__((ext_vector_type(8))) float v8f;

#define EPS 1e-5f

// ---------------------------------------------------------------------------
// WMMA f32 GEMM: C[M x N] = A[M x K] @ W[K x N]  (+bias, optional ReLU)
// One wave (32 threads) computes a 16-row x 64-col tile.
// grid = (M/16, N/64), block = 32.  Requires M%16==0, N%64==0, K%4==0.
// ---------------------------------------------------------------------------
__global__ void gemm_f32_wmma(const float* __restrict__ A,
                              const float* __restrict__ W,
                              float* __restrict__ C,
                              int K, int N, int ldA, int ldC,
                              const float* __restrict__ bias, int relu)
{
    const int lane = threadIdx.x;       // 0..31
    const int half = lane >> 4;         // 0 or 1
    const int l16  = lane & 15;
    const int row0 = blockIdx.x * 16;
    const int col0 = blockIdx.y * 64;
    const int arow = row0 + l16;

    v8f acc0 = {}, acc1 = {}, acc2 = {}, acc3 = {};

    for (int k = 0; k < K; k += 4) {
        const int kk = k + half * 2;
        v2f a;
        a.x = A[(size_t)arow * ldA + kk];
        a.y = A[(size_t)arow * ldA + kk + 1];

        v2f b0, b1, b2, b3;
        const size_t wr0 = (size_t)kk * N;
        const size_t wr1 = (size_t)(kk + 1) * N;
        b0.x = W[wr0 + col0 +  0 + l16];  b0.y = W[wr1 + col0 +  0 + l16];
        b1.x = W[wr0 + col0 + 16 + l16];  b1.y = W[wr1 + col0 + 16 + l16];
        b2.x = W[wr0 + col0 + 32 + l16];  b2.y = W[wr1 + col0 + 32 + l16];
        b3.x = W[wr0 + col0 + 48 + l16];  b3.y = W[wr1 + col0 + 48 + l16];

        acc0 = __builtin_amdgcn_wmma_f32_16x16x4_f32(false, a, false, b0, (short)0, acc0, false, false);
        acc1 = __builtin_amdgcn_wmma_f32_16x16x4_f32(false, a, false, b1, (short)0, acc1, false, false);
        acc2 = __builtin_amdgcn_wmma_f32_16x16x4_f32(false, a, false, b2, (short)0, acc2, false, false);
        acc3 = __builtin_amdgcn_wmma_f32_16x16x4_f32(false, a, false, b3, (short)0, acc3, false, false);
    }

    v8f accs[4] = {acc0, acc1, acc2, acc3};
#pragma unroll
    for (int j = 0; j < 4; ++j) {
        const int col = col0 + j * 16 + l16;
        const float bv = bias ? bias[col] : 0.0f;
#pragma unroll
        for (int v = 0; v < 8; ++v) {
            const int r = row0 + v + half * 8;   // C/D layout: VGPR v -> M=v (+8 hi half)
            float x = accs[j][v] + bv;
            if (relu) x = fmaxf(x, 0.0f);
            C[(size_t)r * ldC + col] = x;
        }
    }
}

// ---------------------------------------------------------------------------
// Degree / norm
// ---------------------------------------------------------------------------
__global__ void deg_kernel(const int* __restrict__ dst, int nE, float* __restrict__ deg)
{
    int i = blockIdx.x * blockDim.x + threadIdx.x;
    if (i < nE) atomicAdd(&deg[dst[i]], 1.0f);
}

__global__ void norm_kernel(float* __restrict__ deg, int n)
{
    int i = blockIdx.x * blockDim.x + threadIdx.x;
    if (i < n) {
        float d = deg[i] + 1.0f;              // +1 self loop
        deg[i] = rsqrtf(fmaxf(d, 1.0f));
    }
}

// ---------------------------------------------------------------------------
// Aggregation: hout[i] = bias + norm[i]^2 * hw[i]  (self loop), then edges add
// ---------------------------------------------------------------------------
__global__ void init_out_kernel(const float* __restrict__ hw,
                                const float* __restrict__ norm,
                                const float* __restrict__ bias,
                                float* __restrict__ hout, int n)
{
    int t = blockIdx.x * blockDim.x + threadIdx.x;
    if (t < n * 64) {
        int i = t >> 6, ch = t & 63;
        float nn = norm[i];
        hout[t] = bias[ch] + nn * nn * hw[t];
    }
}

__global__ void scatter_kernel(const int* __restrict__ src,
                               const int* __restrict__ dst,
                               const float* __restrict__ norm,
                               const float* __restrict__ hw,
                               float* __restrict__ hout, int nE)
{
    int t = blockIdx.x * blockDim.x + threadIdx.x;
    int e = t >> 6, ch = t & 63;
    if (e < nE) {
        int s = src[e], d = dst[e];
        float c = norm[s] * norm[d];
        atomicAdd(&hout[(size_t)d * 64 + ch], hw[(size_t)s * 64 + ch] * c);
    }
}

// ---------------------------------------------------------------------------
// BatchNorm over nodes (64 channels): stats then apply into H concat buffer
// ---------------------------------------------------------------------------
__global__ void bn_stats_kernel(const float* __restrict__ h, int n,
                                float* __restrict__ stats /* [128]: sum, sumsq */)
{
    __shared__ float ssum[256], ssq[256];
    const int ch  = threadIdx.x & 63;
    const int sub = threadIdx.x >> 6;     // 0..3
    const int r0  = blockIdx.x * 256;
    const int r1  = (r0 + 256 < n) ? (r0 + 256) : n;
    float sum = 0.f, sq = 0.f;
    for (int r = r0 + sub; r < r1; r += 4) {
        float v = h[(size_t)r * 64 + ch];
        sum += v; sq += v * v;
    }
    ssum[threadIdx.x] = sum; ssq[threadIdx.x] = sq;
    __syncthreads();
    if (sub == 0) {
        sum = ssum[ch] + ssum[64 + ch] + ssum[128 + ch] + ssum[192 + ch];
        sq  = ssq[ch]  + ssq[64 + ch]  + ssq[128 + ch]  + ssq[192 + ch];
        atomicAdd(&stats[ch], sum);
        atomicAdd(&stats[64 + ch], sq);
    }
}

__global__ void bn_apply_kernel(const float* __restrict__ h,
                                const float* __restrict__ stats,
                                const float* __restrict__ g,
                                const float* __restrict__ b,
                                float* __restrict__ H, int n, int colOff)
{
    int t = blockIdx.x * blockDim.x + threadIdx.x;
    if (t < n * 64) {
        int i = t >> 6, ch = t & 63;
        float invN = 1.0f / (float)n;
        float mean = stats[ch] * invN;
        float var  = stats[64 + ch] * invN - mean * mean;
        float r    = rsqrtf(var + EPS);
        H[(size_t)i * 256 + colOff + ch] = (h[t] - mean) * r * g[ch] + b[ch];
    }
}

// ---------------------------------------------------------------------------
// Pooling: per-graph add / count / max (monotone uint key trick for fp max)
// ---------------------------------------------------------------------------
__device__ __forceinline__ unsigned fkey(float f)
{
    unsigned b = __float_as_uint(f);
    return (b & 0x80000000u) ? ~b : (b | 0x80000000u);
}
__device__ __forceinline__ float funkey(unsigned k)
{
    return (k & 0x80000000u) ? __uint_as_float(k & 0x7FFFFFFFu) : __uint_as_float(~k);
}

__global__ void pool_init_kernel(float* __restrict__ add, unsigned* __restrict__ mkey,
                                 float* __restrict__ cnt)
{
    int t = blockIdx.x * blockDim.x + threadIdx.x;
    if (t < 512 * 256) {
        add[t]  = 0.f;
        mkey[t] = 0x007FFFFFu;               // key(-inf)
        if (t < 512) cnt[t] = 0.f;
    }
}

__global__ void pool_scatter_kernel(const float* __restrict__ H,
                                    const int* __restrict__ batch,
                                    float* __restrict__ add,
                                    unsigned* __restrict__ mkey,
                                    float* __restrict__ cnt, int n)
{
    int t = blockIdx.x * blockDim.x + threadIdx.x;
    if (t < n * 256) {
        int i = t >> 8, ch = t & 255;
        int g = batch[i];
        float v = H[t];
        atomicAdd(&add[(size_t)g * 256 + ch], v);
        atomicMax(&mkey[(size_t)g * 256 + ch], fkey(v));
        if (ch == 0) atomicAdd(&cnt[g], 1.0f);
    }
}

__global__ void pool_final_kernel(const float* __restrict__ add,
                                  const unsigned* __restrict__ mkey,
                                  const float* __restrict__ cnt,
                                  float* __restrict__ Hg)
{
    int t = blockIdx.x * blockDim.x + threadIdx.x;
    if (t < 512 * 256) {
        int g = t >> 8, ch = t & 255;
        float c  = cnt[g];
        float a  = add[t];
        float avg = a / fmaxf(c, 1.0f);
        float mx  = (c > 0.f) ? funkey(mkey[t]) : 0.f;
        Hg[(size_t)g * 768 + ch]        = avg;
        Hg[(size_t)g * 768 + 256 + ch]  = a;
        Hg[(size_t)g * 768 + 512 + ch]  = mx;
    }
}

// ---------------------------------------------------------------------------
// Column BN over Hg [512 x 768]: one block per column
// ---------------------------------------------------------------------------
__global__ void bn_cols_kernel(const float* __restrict__ Hg,
                               const float* __restrict__ g,
                               const float* __restrict__ b,
                               float* __restrict__ Z)
{
    __shared__ float ssum[256], ssq[256];
    const int j = blockIdx.x;
    const int tid = threadIdx.x;
    float v0 = Hg[(size_t)tid * 768 + j];
    float v1 = Hg[(size_t)(tid + 256) * 768 + j];
    ssum[tid] = v0 + v1;
    ssq[tid]  = v0 * v0 + v1 * v1;
    __syncthreads();
    for (int s = 128; s > 0; s >>= 1) {
        if (tid < s) { ssum[tid] += ssum[tid + s]; ssq[tid] += ssq[tid + s]; }
        __syncthreads();
    }
    float mean = ssum[0] * (1.0f / 512.0f);
    float var  = ssq[0] * (1.0f / 512.0f) - mean * mean;
    float r    = rsqrtf(var + EPS);
    float gg = g[j], bb = b[j];
    Z[(size_t)tid * 768 + j]         = (v0 - mean) * r * gg + bb;
    Z[(size_t)(tid + 256) * 768 + j] = (v1 - mean) * r * gg + bb;
}

// ---------------------------------------------------------------------------
// Head: out = log_softmax(z2 @ w3 + b3), one wave per row
// ---------------------------------------------------------------------------
__global__ void head_kernel(const float* __restrict__ z2,
                            const float* __restrict__ w3,
                            const float* __restrict__ b3,
                            float* __restrict__ out)
{
    const int row = blockIdx.x;
    const int lane = threadIdx.x;
    float p[10];
#pragma unroll
    for (int c = 0; c < 10; ++c) p[c] = 0.f;
    for (int k = lane; k < 256; k += 32) {
        float z = z2[(size_t)row * 256 + k];
#pragma unroll
        for (int c = 0; c < 10; ++c) p[c] += z * w3[(size_t)k * 10 + c];
    }
#pragma unroll
    for (int c = 0; c < 10; ++c)
        for (int off = 16; off > 0; off >>= 1)
            p[c] += __shfl_down(p[c], off, 32);
    if (lane == 0) {
        float m = -3.4e38f;
#pragma unroll
        for (int c = 0; c < 10; ++c) { p[c] += b3[c]; m = fmaxf(m, p[c]); }
        float s = 0.f;
#pragma unroll
        for (int c = 0; c < 10; ++c) s += __expf(p[c] - m);
        float ls = __logf(s);
#pragma unroll
        for (int c = 0; c < 10; ++c) out[(size_t)row * 10 + c] = p[c] - m - ls;
    }
}

// ---------------------------------------------------------------------------
// Launch
// ---------------------------------------------------------------------------
extern "C" void kernel_launch(void* const* d_in, const int* in_sizes, int n_in,
                              void* d_out, int out_size, void* d_ws, size_t ws_size,
                              hipStream_t stream)
{
    const float* x     = (const float*)d_in[0];
    const int*   src   = (const int*)  d_in[1];
    const int*   dst   = (const int*)  d_in[2];
    const int*   batch = (const int*)  d_in[3];
    const float* W0    = (const float*)d_in[4];
    const float* Ws    = (const float*)d_in[5];
    const float* bs    = (const float*)d_in[6];
    const float* g_bn  = (const float*)d_in[7];
    const float* b_bn  = (const float*)d_in[8];
    const float* g_out = (const float*)d_in[9];
    const float* b_out = (const float*)d_in[10];
    const float* w1    = (const float*)d_in[11];
    const float* b1    = (const float*)d_in[12];
    const float* w2    = (const float*)d_in[13];
    const float* b2    = (const float*)d_in[14];
    const float* w3    = (const float*)d_in[15];
    const float* b3    = (const float*)d_in[16];

    const int N = in_sizes[0] / 128;   // 100000
    const int E = in_sizes[1];         // 1600000

    // workspace carve-out
    char* p = (char*)d_ws;
    auto alloc = [&](size_t bytes) -> void* {
        void* r = (void*)p;
        p += (bytes + 255) & ~(size_t)255;
        return r;
    };
    float*    norm  = (float*)   alloc((size_t)N * 4);
    float*    hw    = (float*)   alloc((size_t)N * 64 * 4);
    float*    hout  = (float*)   alloc((size_t)N * 64 * 4);
    float*    H     = (float*)   alloc((size_t)N * 256 * 4);
    float*    stats = (float*)   alloc(128 * 4);
    float*    addb  = (float*)   alloc((size_t)512 * 256 * 4);
    unsigned* mkey  = (unsigned*)alloc((size_t)512 * 256 * 4);
    float*    cnt   = (float*)   alloc(512 * 4);
    float*    Hg    = (float*)   alloc((size_t)512 * 768 * 4);
    float*    Z     = (float*)   alloc((size_t)512 * 768 * 4);
    float*    z1    = (float*)   alloc((size_t)512 * 512 * 4);
    float*    z2    = (float*)   alloc((size_t)512 * 256 * 4);

    const int T = 256;

    // degrees + symmetric norm
    hipMemsetAsync(norm, 0, (size_t)N * 4, stream);
    deg_kernel <<<(E + T - 1) / T, T, 0, stream>>>(dst, E, norm);
    norm_kernel<<<(N + T - 1) / T, T, 0, stream>>>(norm, N);

    // 4 GCN layers
    for (int l = 0; l < 4; ++l) {
        const float* Wl  = (l == 0) ? W0 : (Ws + (size_t)(l - 1) * 64 * 64);
        const int    Kl  = (l == 0) ? 128 : 64;
        const float* Ain = (l == 0) ? x : (H + (size_t)(l - 1) * 64);
        const int    ldA = (l == 0) ? 128 : 256;

        gemm_f32_wmma<<<dim3(N / 16, 1), 32, 0, stream>>>(
            Ain, Wl, hw, Kl, 64, ldA, 64, nullptr, 0);

        init_out_kernel<<<((size_t)N * 64 + T - 1) / T, T, 0, stream>>>(
            hw, norm, bs + (size_t)l * 64, hout, N);
        scatter_kernel<<<((size_t)E * 64 + T - 1) / T, T, 0, stream>>>(
            src, dst, norm, hw, hout, E);

        hipMemsetAsync(stats, 0, 128 * 4, stream);
        bn_stats_kernel<<<(N + 255) / 256, 256, 0, stream>>>(hout, N, stats);
        bn_apply_kernel<<<((size_t)N * 64 + T - 1) / T, T, 0, stream>>>(
            hout, stats, g_bn + (size_t)l * 64, b_bn + (size_t)l * 64, H, N, l * 64);
    }

    // pooling
    pool_init_kernel   <<<(512 * 256 + T - 1) / T, T, 0, stream>>>(addb, mkey, cnt);
    pool_scatter_kernel<<<((size_t)N * 256 + T - 1) / T, T, 0, stream>>>(H, batch, addb, mkey, cnt, N);
    pool_final_kernel  <<<(512 * 256 + T - 1) / T, T, 0, stream>>>(addb, mkey, cnt, Hg);

    // readout
    bn_cols_kernel<<<768, 256, 0, stream>>>(Hg, g_out, b_out, Z);
    gemm_f32_wmma<<<dim3(512 / 16, 512 / 64), 32, 0, stream>>>(Z,  w1, z1, 768, 512, 768, 512, b1, 1);
    gemm_f32_wmma<<<dim3(512 / 16, 256 / 64), 32, 0, stream>>>(z1, w2, z2, 512, 256, 512, 256, b2, 1);
    head_kernel<<<512, 32, 0, stream>>>(z2, w3, b3, (float*)d_out);
}